// AlternateConvolution_55989193670681
// MI455X (gfx1250) — compile-verified
//
#include <hip/hip_runtime.h>

// B=2, N=1024, E=8192, feature dims 64.
#define BN 1024
#define EE 8192
#define DD 64

typedef float v2f __attribute__((ext_vector_type(2)));
typedef float v8f __attribute__((ext_vector_type(8)));

__device__ __forceinline__ v8f wmma_f32(v2f a, v2f b, v8f c) {
    // V_WMMA_F32_16X16X4_F32: D = A(16x4) * B(4x16) + C(16x16)
    return __builtin_amdgcn_wmma_f32_16x16x4_f32(
        false, a, false, b, (short)0, c, false, false);
}

// ---------------------------------------------------------------------------
// Kernel 1: scale[b*E+e] = dot(H_e[b*E+e, 0:64], p[0:64])
// One wave (32 lanes) per row; each lane takes 2 elements; shfl_xor reduce.
// ---------------------------------------------------------------------------
__global__ __launch_bounds__(256)
void scale_kernel(const float* __restrict__ He, const float* __restrict__ p,
                  float* __restrict__ scale, int rows) {
    int gwave = (int)((blockIdx.x * blockDim.x + threadIdx.x) >> 5);
    int lane  = (int)(threadIdx.x & 31);
    if (gwave >= rows) return;
    const float* r = He + (size_t)gwave * 64;
    float s = r[lane] * p[lane] + r[lane + 32] * p[lane + 32];
#pragma unroll
    for (int off = 16; off > 0; off >>= 1) s += __shfl_xor(s, off, 32);
    if (lane == 0) scale[gwave] = s;
}

// ---------------------------------------------------------------------------
// Kernel 2: HW[r,d] = sum_k H_v[r,k] * W[k,d]   (2048x64 @ 64x64)
// One wave per 16-row tile; 4 column tiles of 16; K=64 in steps of 4 via WMMA.
// ---------------------------------------------------------------------------
__global__ __launch_bounds__(256)
void hw_kernel(const float* __restrict__ Hv, const float* __restrict__ W,
               float* __restrict__ HW) {
    int lane = (int)(threadIdx.x & 31);
    int wv   = (int)(threadIdx.x >> 5);
    int rt   = (int)blockIdx.x * 8 + wv;        // 128 row tiles
    int r0   = rt * 16;
    int hi   = lane >> 4;                       // half-wave: k split
    int ln   = lane & 15;

    v8f c[4] = {v8f{}, v8f{}, v8f{}, v8f{}};
    const float* pA = Hv + (size_t)(r0 + ln) * DD + hi * 2;

    for (int k0 = 0; k0 < DD; k0 += 4) {
        v2f a = *(const v2f*)(pA + k0);
        int kr = k0 + hi * 2;
#pragma unroll
        for (int t = 0; t < 4; ++t) {
            v2f b;
            b.x = W[(size_t)kr * DD + t * 16 + ln];
            b.y = W[(size_t)(kr + 1) * DD + t * 16 + ln];
            c[t] = wmma_f32(a, b, c[t]);
        }
    }
#pragma unroll
    for (int t = 0; t < 4; ++t) {
#pragma unroll
        for (int r = 0; r < 8; ++r) {
            int M = r + hi * 8;
            HW[(size_t)(r0 + M) * DD + t * 16 + ln] = c[t][r];
        }
    }
}

// ---------------------------------------------------------------------------
// Kernel 3 (fused): for each (batch, 16-row n-tile):
//   S[n,m] = sum_k T[n,k]*scale[k]*T[m,k]   (K = 8192, fp32 WMMA)
//   A[n,m] = (n==m ? 1 : S[n,m]) * adj_v[n,m]
//   out[n,0:64] = sum_m A[n,m] * HW[m,0:64] + bias
// 8 waves; each wave owns 8 m-tiles concurrently so the scaled A-fragment is
// amortized over 8 WMMAs per k-step. Second GEMM chained via LDS reshape.
// ---------------------------------------------------------------------------
__global__ __launch_bounds__(256)
void fused_kernel(const float* __restrict__ T, const float* __restrict__ scale,
                  const float* __restrict__ adjv, const float* __restrict__ HW,
                  const float* __restrict__ bias, float* __restrict__ out) {
    __shared__ float lds_acc[16 * DD];          // cross-wave output reduction
    __shared__ float lds_S[8][256];             // per-wave 16x16 staging

    int batch = (int)blockIdx.x >> 6;
    int nt    = (int)blockIdx.x & 63;
    int n0    = nt * 16;

    int lane = (int)(threadIdx.x & 31);
    int wv   = (int)(threadIdx.x >> 5);
    int hi   = lane >> 4;
    int ln   = lane & 15;

    for (int idx = (int)threadIdx.x; idx < 16 * DD; idx += 256) lds_acc[idx] = 0.0f;
    __syncthreads();

    const float* Tb = T + (size_t)batch * BN * EE;
    const float* pA = Tb + (size_t)(n0 + ln) * EE + hi * 2;
    const float* pS = scale + (size_t)batch * EE + hi * 2;

    const float* pB[8];
#pragma unroll
    for (int t = 0; t < 8; ++t) {
        int m0 = (t * 8 + wv) * 16;
        pB[t] = Tb + (size_t)(m0 + ln) * EE + hi * 2;
    }

    v8f S[8];
#pragma unroll
    for (int t = 0; t < 8; ++t) S[t] = v8f{};

    // ---- main K loop: 2048 WMMA k-steps, 8 tiles each ----
    for (int k = 0; k < EE; k += 4) {
        v2f a  = *(const v2f*)(pA + k);
        v2f sc = *(const v2f*)(pS + k);
        a.x *= sc.x;
        a.y *= sc.y;
#pragma unroll
        for (int t = 0; t < 8; ++t) {
            v2f b = *(const v2f*)(pB[t] + k);
            S[t] = wmma_f32(a, b, S[t]);
        }
    }

    // ---- epilogue: diag/adj_v, then chained GEMM against HW ----
    float* myS = &lds_S[wv][0];
    v8f Out[4] = {v8f{}, v8f{}, v8f{}, v8f{}};
    const float* adjb = adjv + (size_t)batch * BN * BN;

    for (int t = 0; t < 8; ++t) {
        int m0 = (t * 8 + wv) * 16;
        int m  = m0 + ln;
#pragma unroll
        for (int r = 0; r < 8; ++r) {
            int M = r + hi * 8;
            int n = n0 + M;
            float v = S[t][r];
            if (n == m) v = 1.0f;
            v *= adjb[(size_t)n * BN + m];
            myS[M * 16 + ln] = v;             // C layout -> LDS (in-order DS)
        }
        __builtin_amdgcn_wave_barrier();
        // Out[dt] += Sadj(16x16) @ HW[m0:m0+16, dt*16:dt*16+16]
#pragma unroll
        for (int kk = 0; kk < 16; kk += 4) {
            v2f a2 = *(const v2f*)(myS + ln * 16 + kk + hi * 2);
            int krow = batch * BN + m0 + kk + hi * 2;
#pragma unroll
            for (int dt = 0; dt < 4; ++dt) {
                v2f b2;
                b2.x = HW[(size_t)krow * DD + dt * 16 + ln];
                b2.y = HW[(size_t)(krow + 1) * DD + dt * 16 + ln];
                Out[dt] = wmma_f32(a2, b2, Out[dt]);
            }
        }
        __builtin_amdgcn_wave_barrier();
    }

    // ---- cross-wave reduction (ds_add_f32) ----
#pragma unroll
    for (int dt = 0; dt < 4; ++dt) {
#pragma unroll
        for (int r = 0; r < 8; ++r) {
            int M = r + hi * 8;
            atomicAdd(&lds_acc[M * DD + dt * 16 + ln], Out[dt][r]);
        }
    }
    __syncthreads();

    for (int idx = (int)threadIdx.x; idx < 16 * DD; idx += 256) {
        int row = idx >> 6;
        int d   = idx & 63;
        out[((size_t)batch * BN + n0 + row) * DD + d] = lds_acc[idx] + bias[d];
    }
}

// ---------------------------------------------------------------------------
extern "C" void kernel_launch(void* const* d_in, const int* in_sizes, int n_in,
                              void* d_out, int out_size, void* d_ws, size_t ws_size,
                              hipStream_t stream) {
    (void)in_sizes; (void)n_in; (void)out_size; (void)ws_size;
    const float* Hv   = (const float*)d_in[0];   // [2048,64]
    const float* He   = (const float*)d_in[1];   // [16384,64]
    // d_in[2] = adj_e : never read by the reference
    const float* adjv = (const float*)d_in[3];   // [2048,1024]
    const float* T    = (const float*)d_in[4];   // [2048,8192]
    const float* W    = (const float*)d_in[5];   // [64,64]
    const float* p    = (const float*)d_in[6];   // [64]
    const float* bias = (const float*)d_in[7];   // [64]
    float* out = (float*)d_out;

    float* scale = (float*)d_ws;                 // 2*8192 floats
    float* HW    = scale + 2 * EE;               // 2048*64 floats (ws: 576 KB)

    scale_kernel<<<2048, 256, 0, stream>>>(He, p, scale, 2 * EE);
    hw_kernel<<<16, 256, 0, stream>>>(Hv, W, HW);
    fused_kernel<<<128, 256, 0, stream>>>(T, scale, adjv, HW, bias, out);

    // Second tuple output: H_e passthrough.
    hipMemcpyAsync(out + (size_t)2 * BN * DD, He,
                   (size_t)2 * EE * 64 * sizeof(float),
                   hipMemcpyDeviceToDevice, stream);
}